// PtModule_76166950027845
// MI455X (gfx1250) — compile-verified
//
#include <hip/hip_runtime.h>
#include <hip/hip_bf16.h>

// Reference: x:[65536,1024] f32 ; y = 2x ; rowsum = sum(y) ; out = y - (rowsum>10 ? 1 : 0)
// Pure HBM streaming (512 MB total @ 23.3 TB/s => ~22us floor). No matmul => no WMMA.
// CDNA5 path: GLOBAL_LOAD_ASYNC_TO_LDS_B128 (ASYNCcnt) with per-wave double buffering,
// th:TH_LOAD_NT so the read-once 256MB stream doesn't thrash the 192MB L2,
// two-pass LDS reads, wave32 shuffle reduction, nontemporal b128 stores.

typedef float v4f __attribute__((ext_vector_type(4)));

#define ROW_ELEMS 1024
#define ROW_BYTES 4096
#define WAVES_PER_BLOCK 8
#define BLOCK_THREADS (WAVES_PER_BLOCK * 32)
#define CHUNKS 8 // 8 chunks * (32 lanes * 16B) = 4096B = one row
#define MAX_BLOCKS 2048

// Issue one row (4KB) as 8 async b128 global->LDS transfers (non-temporal in L2).
// vdst operand = per-lane LDS byte offset (low 32 bits of the flat shared address),
// vaddr operand = 64-bit global address, saddr = off.
__device__ __forceinline__ void issue_row_async(const float* __restrict__ x,
                                                int row,
                                                const float* ldsBuf,
                                                int lane) {
  unsigned long long g = (unsigned long long)(uintptr_t)x
                       + (unsigned long long)row * ROW_BYTES
                       + (unsigned)lane * 16u;
  unsigned l = (unsigned)(uintptr_t)ldsBuf + (unsigned)lane * 16u;
#pragma unroll
  for (int k = 0; k < CHUNKS; ++k) {
    asm volatile("global_load_async_to_lds_b128 %0, %1, off th:TH_LOAD_NT"
                 :
                 : "v"(l), "v"(g)
                 : "memory");
    g += 512ull; // 32 lanes * 16B
    l += 512u;
  }
}

__global__ __launch_bounds__(BLOCK_THREADS)
void row_double_condsub_kernel(const float* __restrict__ x,
                               float* __restrict__ out,
                               int nrows) {
  // 8 waves * 2 buffers * 4KB = 64KB LDS per workgroup (<= 320KB/WGP)
  __shared__ float lds[WAVES_PER_BLOCK * 2 * ROW_ELEMS];

  const int lane       = (int)(threadIdx.x & 31u);
  const int wave       = (int)(threadIdx.x >> 5);
  const int waveGlobal = (int)blockIdx.x * WAVES_PER_BLOCK + wave;
  const int waveStride = (int)gridDim.x * WAVES_PER_BLOCK;

  float* buf0 = &lds[(wave * 2 + 0) * ROW_ELEMS];
  float* buf1 = &lds[(wave * 2 + 1) * ROW_ELEMS];

  if (waveGlobal >= nrows) return; // wave-uniform exit; active waves keep EXEC all-1s

  // Prime the pipeline: fetch first row into buffer 0.
  issue_row_async(x, waveGlobal, buf0, lane);

  int i = 0;
  for (int row = waveGlobal; row < nrows; row += waveStride, ++i) {
    const int next = row + waveStride;
    const float* __restrict__ buf = (i & 1) ? buf1 : buf0;

    if (next < nrows) {
      // Prefetch next row into the other buffer, then wait only for the
      // current row's 8 transfers (<=8 outstanding == next row still in flight).
      issue_row_async(x, next, (i & 1) ? buf0 : buf1, lane);
      asm volatile("s_wait_asynccnt 0x8" ::: "memory");
    } else {
      asm volatile("s_wait_asynccnt 0x0" ::: "memory");
    }

    // Pass 1: per-lane partial sum of x over this row (lane owns 8 float4s).
    float partial = 0.0f;
#pragma unroll
    for (int k = 0; k < CHUNKS; ++k) {
      v4f v = *(const v4f*)(buf + k * 128 + lane * 4);
      partial += (v.x + v.y) + (v.z + v.w);
    }

    // Wave32 butterfly reduction.
#pragma unroll
    for (int off = 16; off >= 1; off >>= 1)
      partial += __shfl_xor(partial, off, 32);

    const float rowsum = 2.0f * partial;           // sum(2x) == 2*sum(x) exactly
    const float sub    = (rowsum > 10.0f) ? 1.0f : 0.0f;

    // Pass 2: y = 2x - sub, streamed out with nontemporal b128 stores.
    float* __restrict__ orow = out + (size_t)row * ROW_ELEMS;
#pragma unroll
    for (int k = 0; k < CHUNKS; ++k) {
      v4f v = *(const v4f*)(buf + k * 128 + lane * 4);
      v4f y = v * 2.0f - sub;
      __builtin_nontemporal_store(y, (v4f*)(orow + k * 128 + lane * 4));
    }
    // Data dependency (ds_load -> nontemporal store) guarantees this buffer is
    // fully read before the next iteration's async writes can land in it.
    // No barriers needed: LDS regions and ASYNCcnt are per-wave private.
  }
}

extern "C" void kernel_launch(void* const* d_in, const int* in_sizes, int n_in,
                              void* d_out, int out_size, void* d_ws, size_t ws_size,
                              hipStream_t stream) {
  const float* x = (const float*)d_in[0];
  float* out     = (float*)d_out;

  const int total = in_sizes[0];          // 65536 * 1024
  const int nrows = total / ROW_ELEMS;    // 65536

  int blocks = (nrows + WAVES_PER_BLOCK - 1) / WAVES_PER_BLOCK;
  if (blocks > MAX_BLOCKS) blocks = MAX_BLOCKS;

  hipLaunchKernelGGL(row_double_condsub_kernel,
                     dim3(blocks), dim3(BLOCK_THREADS), 0, stream,
                     x, out, nrows);
}